// ETSModule_59906203845011
// MI455X (gfx1250) — compile-verified
//
#include <hip/hip_runtime.h>
#include <stdint.h>

// ---------------------------------------------------------------------------
// ETS "AAA" scan, N=4096 series x T=4096 steps, P=12 seasonal.
// One wave32 per 32 series; double-buffered async global->LDS tile staging
// (CDNA5 GLOBAL_LOAD_ASYNC_TO_LDS_B128 / ASYNCcnt), seasonal ring held in
// VGPRs via a 48-step (multiple of lcm(4,12)) fully unrolled body with static indices.
// ---------------------------------------------------------------------------

namespace {
constexpr int PP = 12;      // seasonal period
constexpr int TK = 48;      // timesteps per tile (multiple of 4 and 12)
constexpr int QT = TK / 4;  // 12 float4 quads per row per tile
constexpr int BS = 32;      // series per block == threads per block (one wave32)
constexpr int RS = TK + 4;  // LDS row stride in dwords (208 B: 16B aligned, 2-way banks)
}  // namespace

// ---- CDNA5 async global->LDS copy (ASYNCcnt) ------------------------------
// Probe-confirmed signature: first param is AS(1) pointer to a 16B int vector
// (non-const); second is the LDS destination. Cast accordingly.
typedef int v4i __attribute__((vector_size(4 * sizeof(int))));
typedef __attribute__((address_space(1))) v4i* gptr_b128;
typedef __attribute__((address_space(3))) v4i* lptr_b128;

#if defined(__HIP_DEVICE_COMPILE__) && \
    __has_builtin(__builtin_amdgcn_global_load_async_to_lds_b128)
#define ASYNC_LOAD_B128(g, l)                                     \
  __builtin_amdgcn_global_load_async_to_lds_b128(                 \
      (gptr_b128)(g), (lptr_b128)(l), 0, 0)
#else
// GV mode: VDST = LDS byte address (low 32 bits of generic LDS pointer),
// VADDR = 64-bit global address, saddr = off.
#define ASYNC_LOAD_B128(g, l)                                     \
  asm volatile("global_load_async_to_lds_b128 %0, %1, off"        \
               :: "v"((uint32_t)(uintptr_t)(l)),                  \
                  "v"((uint64_t)(uintptr_t)(g))                   \
               : "memory")
#endif

#if defined(__HIP_DEVICE_COMPILE__) && \
    __has_builtin(__builtin_amdgcn_s_wait_asynccnt)
#define WAIT_ASYNCCNT(n)                          \
  do {                                            \
    __builtin_amdgcn_s_wait_asynccnt(n);          \
    asm volatile("" ::: "memory");                \
  } while (0)
#else
#define WAIT_ASYNCCNT(n) asm volatile("s_wait_asynccnt %0" :: "n"(n) : "memory")
#endif

// ---- scan step (matches reference ETS-AAA algebra exactly) ----------------
// current   = level + trend + s
// y         = current + 0.1*on
// level'    = (level+trend) + alpha*(s+e)
// trend'    = trend + beta*alpha*(s+e)
// s'        = s + gamma*(e - s)
__device__ __forceinline__ void ets_step(int idx, float e, float on,
                                         float (&sea)[PP], float& level,
                                         float& trend, float a, float b,
                                         float g, float& y) {
  float s  = sea[idx];
  float lt = level + trend;
  y        = fmaf(0.1f, on, lt + s);
  float q  = a * (s + e);
  level    = lt + q;
  trend    = fmaf(b, q, trend);
  sea[idx] = fmaf(g, e - s, s);
}

template <int NQ>
__device__ __forceinline__ void compute_tile(const float* __restrict__ eb,
                                             const float* __restrict__ ob,
                                             float* __restrict__ outp,
                                             float (&sea)[PP], float& level,
                                             float& trend, float a, float b,
                                             float g) {
#pragma unroll
  for (int gq = 0; gq < NQ; ++gq) {
    float4 e4 = *(const float4*)(eb + gq * 4);  // ds_load_b128
    float4 o4 = *(const float4*)(ob + gq * 4);  // ds_load_b128
    float4 y4;
    // tile length is a multiple of 12 -> indices are compile-time after unroll
    ets_step((gq * 4 + 0) % PP, e4.x, o4.x, sea, level, trend, a, b, g, y4.x);
    ets_step((gq * 4 + 1) % PP, e4.y, o4.y, sea, level, trend, a, b, g, y4.y);
    ets_step((gq * 4 + 2) % PP, e4.z, o4.z, sea, level, trend, a, b, g, y4.z);
    ets_step((gq * 4 + 3) % PP, e4.w, o4.w, sea, level, trend, a, b, g, y4.w);
    *(float4*)(outp + gq * 4) = y4;             // global_store_b128
  }
}

__global__ __launch_bounds__(BS, 1) void ets_scan_kernel(
    const float* __restrict__ alpha, const float* __restrict__ beta,
    const float* __restrict__ gam, const float* __restrict__ init_level,
    const float* __restrict__ init_trend,
    const float* __restrict__ init_seasonal, const float* __restrict__ obs,
    const float* __restrict__ err, float* __restrict__ out, int T) {
  __shared__ __align__(16) float eT[2][BS * RS];
  __shared__ __align__(16) float oT[2][BS * RS];

  const int tid    = threadIdx.x;
  const int s0     = blockIdx.x * BS;
  const int series = s0 + tid;

  const float a = alpha[series];
  const float b = beta[series];
  const float g = gam[series];
  float level   = init_level[series];
  float trend   = init_trend[series];
  float sea[PP];
#pragma unroll
  for (int j = 0; j < PP; ++j) sea[j] = init_seasonal[series * PP + j];

  // Cooperative coalesced tile loading: 12 async instrs x 32 lanes cover
  // 32 rows x 12 quads; lane m*32+tid -> (row = id/12, quad = id%12) so
  // consecutive lanes fetch consecutive 16B quads within a row.
  uint32_t gofs[QT], lofs[QT];
#pragma unroll
  for (int m = 0; m < QT; ++m) {
    int id  = m * BS + tid;
    int row = id / QT;
    int q   = id - row * QT;
    gofs[m] = (uint32_t)(row * T + q * 4) * 4u;   // bytes within 32-row slab
    lofs[m] = (uint32_t)(row * RS + q * 4) * 4u;  // bytes within LDS tile
  }

  const char* errB = (const char*)(err + (size_t)s0 * T);
  const char* obsB = (const char*)(obs + (size_t)s0 * T);
  float* outRow    = out + (size_t)series * T;

  auto issue_full = [&](int buf, int t0) {
    uint32_t toff = (uint32_t)t0 * 4u;
    char* el = (char*)&eT[buf][0];
    char* ol = (char*)&oT[buf][0];
#pragma unroll
    for (int m = 0; m < QT; ++m) {
      ASYNC_LOAD_B128(errB + (gofs[m] + toff), el + lofs[m]);
      ASYNC_LOAD_B128(obsB + (gofs[m] + toff), ol + lofs[m]);
    }
  };

  // Tail tile: 16 timesteps = 32 rows x 4 quads = 4 instrs per array.
  auto issue_tail16 = [&](int buf, int t0) {
    uint32_t toff = (uint32_t)t0 * 4u;
    char* el = (char*)&eT[buf][0];
    char* ol = (char*)&oT[buf][0];
    int q = tid & 3;
#pragma unroll
    for (int m = 0; m < 4; ++m) {
      int row     = m * (BS / 4) + (tid >> 2);
      uint32_t go = (uint32_t)(row * T + q * 4) * 4u + toff;
      uint32_t lo = (uint32_t)(row * RS + q * 4) * 4u;
      ASYNC_LOAD_B128(errB + go, el + lo);
      ASYNC_LOAD_B128(obsB + go, ol + lo);
    }
  };

  const int nFull = T / TK;         // 85 for T=4096
  const int tail  = T - nFull * TK; // 16 for T=4096 (supported tails: 0, 16)

  issue_full(0, 0);
  for (int i = 0; i < nFull - 1; ++i) {
    issue_full((i + 1) & 1, (i + 1) * TK);
    // Loads retire in order: any tile-i load pending => counter >= 2*QT+1.
    WAIT_ASYNCCNT(2 * QT);
    compute_tile<QT>(&eT[i & 1][tid * RS], &oT[i & 1][tid * RS],
                     outRow + i * TK, sea, level, trend, a, b, g);
  }
  {  // last full tile (prefetch the 16-step tail: 8 async ops)
    const int i = nFull - 1;
    if (tail == 16) {
      issue_tail16((i + 1) & 1, nFull * TK);
      WAIT_ASYNCCNT(8);
    } else {
      WAIT_ASYNCCNT(0);
    }
    compute_tile<QT>(&eT[i & 1][tid * RS], &oT[i & 1][tid * RS],
                     outRow + i * TK, sea, level, trend, a, b, g);
  }
  if (tail == 16) {  // t0 = 4080 = 12*340 -> seasonal phase 0, static indices
    WAIT_ASYNCCNT(0);
    compute_tile<4>(&eT[nFull & 1][tid * RS], &oT[nFull & 1][tid * RS],
                    outRow + nFull * TK, sea, level, trend, a, b, g);
  }
}

// ---------------------------------------------------------------------------
extern "C" void kernel_launch(void* const* d_in, const int* in_sizes, int n_in,
                              void* d_out, int out_size, void* d_ws,
                              size_t ws_size, hipStream_t stream) {
  (void)n_in; (void)out_size; (void)d_ws; (void)ws_size;
  const float* alpha = (const float*)d_in[0];
  const float* beta  = (const float*)d_in[1];
  const float* gam   = (const float*)d_in[2];
  const float* il    = (const float*)d_in[3];
  const float* itr   = (const float*)d_in[4];
  const float* isea  = (const float*)d_in[5];
  const float* obs   = (const float*)d_in[6];
  const float* err   = (const float*)d_in[7];
  float* out = (float*)d_out;

  const int N = in_sizes[0];          // 4096 series
  const int T = in_sizes[6] / N;      // 4096 timesteps

  dim3 grid(N / BS), block(BS);       // 128 single-wave workgroups
  ets_scan_kernel<<<grid, block, 0, stream>>>(alpha, beta, gam, il, itr, isea,
                                              obs, err, out, T);
}